// MonotonicEnergy_16432544875120
// MI455X (gfx1250) — compile-verified
//
#include <hip/hip_runtime.h>
#include <hip/hip_bf16.h>

typedef __attribute__((ext_vector_type(2))) float v2f;
typedef __attribute__((ext_vector_type(8))) float v8f;

#define ADIM 512
#define HN 4
#define KLEN 512
#define QLEN 64

// ---------------------------------------------------------------------------
// Kernel 1: Y[M, 512] = X[M, 512] @ W[512, 512]^T (+ bias)
// Register-blocked: each wave computes a 16x64 strip (4 N-tiles), so each
// A-fragment load feeds 4 V_WMMA_F32_16X16X4_F32 ops (1.25 loads/WMMA).
// A-frag layout (16x4 f32): lanes 0-15 hold rows M=0..15 at K={0,1},
// lanes 16-31 the same rows at K={2,3}; B-frag (4x16) symmetric with N.
// ---------------------------------------------------------------------------
__global__ __launch_bounds__(256) void proj_gemm(const float* __restrict__ X,
                                                 const float* __restrict__ W,
                                                 const float* __restrict__ bias,
                                                 float* __restrict__ Y,
                                                 int M) {
  const int tid  = threadIdx.x;
  const int lane = tid & 31;
  const int wave = tid >> 5;
  const int tile = blockIdx.x * 8 + wave;     // wave-tile id
  const int mt   = tile >> 3;                 // / (512/64) N-groups
  const int ng   = tile & 7;
  const int m0   = mt * 16;
  const int n0   = ng * 64;                   // 4 consecutive 16-wide N-tiles

  const int rl   = lane & 15;                 // row (A) / col (B) within tile
  const int koff = (lane >> 4) * 2;           // K sub-offset per half-wave

  const float* arow = X + (size_t)(m0 + rl) * ADIM + koff;
  const float* brow = W + (size_t)(n0 + rl) * ADIM + koff;  // +j*16*ADIM per tile

  v8f c0 = {}, c1 = {}, c2 = {}, c3 = {};
#pragma unroll 4
  for (int k0 = 0; k0 < ADIM; k0 += 4) {
    v2f a  = *(const v2f*)(arow + k0);
    v2f b0 = *(const v2f*)(brow + k0);
    v2f b1 = *(const v2f*)(brow + 16 * ADIM + k0);
    v2f b2 = *(const v2f*)(brow + 32 * ADIM + k0);
    v2f b3 = *(const v2f*)(brow + 48 * ADIM + k0);
    c0 = __builtin_amdgcn_wmma_f32_16x16x4_f32(false, a, false, b0, (short)0, c0, false, false);
    c1 = __builtin_amdgcn_wmma_f32_16x16x4_f32(false, a, false, b1, (short)0, c1, false, false);
    c2 = __builtin_amdgcn_wmma_f32_16x16x4_f32(false, a, false, b2, (short)0, c2, false, false);
    c3 = __builtin_amdgcn_wmma_f32_16x16x4_f32(false, a, false, b3, (short)0, c3, false, false);
  }

  // C/D layout: VGPR i holds (M = m0 + i + 8*(lane>=16), N = ntile_base + (lane&15))
  const int nl     = lane & 15;
  const int rowadd = (lane >> 4) * 8;
  v8f acc[4] = {c0, c1, c2, c3};
#pragma unroll
  for (int j = 0; j < 4; ++j) {
    const int   nb = n0 + j * 16;
    const float bv = bias ? bias[nb + nl] : 0.0f;
#pragma unroll
    for (int i = 0; i < 8; ++i) {
      const int m = m0 + i + rowadd;
      Y[(size_t)m * ADIM + nb + nl] = acc[j][i] + bv;
    }
  }
}

// ---------------------------------------------------------------------------
// Kernel 2: one block per (b,q).  E[k,n] = relu(Kp[b,k,:] + Qp[b,q,:]) @ W2^T
// with W2[n,a] = g[n] * v[n,a] / ||v[n,:]||, then +r and mask -> out[b,n,q,k].
// Register-blocked: each wave carries its 4 k-tiles simultaneously so the
// LDS q-row / W2 B-fragment loads are shared across 4 WMMAs per K-step and
// the 4 accumulator chains hide WMMA latency.  N padded 4->16 via lane
// select (no EXEC divergence around WMMA).
// ---------------------------------------------------------------------------
__global__ __launch_bounds__(256) void energy_kernel(
    const float* __restrict__ Kp, const float* __restrict__ Qp,
    const float* __restrict__ vv, const float* __restrict__ vg,
    const float* __restrict__ rscal, const int* __restrict__ mask,
    float* __restrict__ out) {
  __shared__ float s_w2[HN * ADIM];   // 8 KB
  __shared__ float s_q[ADIM];         // 2 KB
  __shared__ float s_part[HN][32];
  __shared__ float s_scale[HN];

  const int tid  = threadIdx.x;
  const int lane = tid & 31;
  const int wave = tid >> 5;
  const int b    = blockIdx.x >> 6;   // / QLEN
  const int q    = blockIdx.x & 63;   // % QLEN

  // row norms of v_v (4 rows x 512)
  if (tid < HN * 32) {
    const int n = tid >> 5, j = tid & 31;
    float s = 0.f;
    for (int a = j; a < ADIM; a += 32) {
      const float v = vv[n * ADIM + a];
      s += v * v;
    }
    s_part[n][j] = s;
  }
  __syncthreads();
  if (tid < HN) {
    float s = 0.f;
#pragma unroll
    for (int j = 0; j < 32; ++j) s += s_part[tid][j];
    s_scale[tid] = vg[tid] / sqrtf(s);
  }
  __syncthreads();
  for (int idx = tid; idx < HN * ADIM; idx += 256)
    s_w2[idx] = vv[idx] * s_scale[idx >> 9];
  for (int idx = tid; idx < ADIM; idx += 256)
    s_q[idx] = Qp[(size_t)(b * QLEN + q) * ADIM + idx];
  __syncthreads();

  const int   rl     = lane & 15;
  const int   koff   = (lane >> 4) * 2;
  const int   bsel   = rl & 3;                   // clamp N lane to 0..3
  const float bscale = (rl < HN) ? 1.0f : 0.0f;  // zero-pad N = 4..15
  const float rr     = rscal[0];

  // 4 k-tiles per wave, processed together: k0 = (wave*4 + j)*16
  const float* krow = Kp + (size_t)(b * KLEN + wave * 64 + rl) * ADIM + koff;

  v8f c0 = {}, c1 = {}, c2 = {}, c3 = {};
#pragma unroll 4
  for (int a0 = 0; a0 < ADIM; a0 += 4) {
    v2f qv = *(const v2f*)(s_q + a0 + koff);                // LDS (shared x4)
    v2f bv = *(const v2f*)(s_w2 + bsel * ADIM + a0 + koff); // LDS (shared x4)
    bv.x *= bscale;
    bv.y *= bscale;
    v2f k0v = *(const v2f*)(krow + a0);
    v2f k1v = *(const v2f*)(krow + 16 * ADIM + a0);
    v2f k2v = *(const v2f*)(krow + 32 * ADIM + a0);
    v2f k3v = *(const v2f*)(krow + 48 * ADIM + a0);
    v2f a0v, a1v, a2v, a3v;
    a0v.x = fmaxf(k0v.x + qv.x, 0.0f); a0v.y = fmaxf(k0v.y + qv.y, 0.0f);
    a1v.x = fmaxf(k1v.x + qv.x, 0.0f); a1v.y = fmaxf(k1v.y + qv.y, 0.0f);
    a2v.x = fmaxf(k2v.x + qv.x, 0.0f); a2v.y = fmaxf(k2v.y + qv.y, 0.0f);
    a3v.x = fmaxf(k3v.x + qv.x, 0.0f); a3v.y = fmaxf(k3v.y + qv.y, 0.0f);
    c0 = __builtin_amdgcn_wmma_f32_16x16x4_f32(false, a0v, false, bv, (short)0, c0, false, false);
    c1 = __builtin_amdgcn_wmma_f32_16x16x4_f32(false, a1v, false, bv, (short)0, c1, false, false);
    c2 = __builtin_amdgcn_wmma_f32_16x16x4_f32(false, a2v, false, bv, (short)0, c2, false, false);
    c3 = __builtin_amdgcn_wmma_f32_16x16x4_f32(false, a3v, false, bv, (short)0, c3, false, false);
  }

  // epilogue: element (k = k0 + i + 8*(lane>=16), n = lane&15), n<4 stored
  const int rowadd = (lane >> 4) * 8;
  v8f acc[4] = {c0, c1, c2, c3};
  if (rl < HN) {                                 // divergence only after WMMA
    const int n = rl;
#pragma unroll
    for (int j = 0; j < 4; ++j) {
      const int k0 = (wave * 4 + j) * 16;
#pragma unroll
      for (int i = 0; i < 8; ++i) {
        const int k = k0 + i + rowadd;
        float val = acc[j][i] + rr;
        if (mask[(size_t)(b * QLEN + q) * KLEN + k] == 0)
          val = -3.402823466e38f;                // float32 min
        out[((size_t)(b * HN + n) * QLEN + q) * KLEN + k] = val;
      }
    }
  }
}

// ---------------------------------------------------------------------------
extern "C" void kernel_launch(void* const* d_in, const int* in_sizes, int n_in,
                              void* d_out, int out_size, void* d_ws, size_t ws_size,
                              hipStream_t stream) {
  const float* key   = (const float*)d_in[0];  // [4, 512, 512]
  const float* query = (const float*)d_in[1];  // [4, 64, 512]
  const int*   mask  = (const int*)d_in[2];    // [4, 64, 512]
  const float* wk    = (const float*)d_in[3];  // [512, 512]
  const float* wkb   = (const float*)d_in[4];  // [512]
  const float* wq    = (const float*)d_in[5];  // [512, 512]
  const float* vv    = (const float*)d_in[6];  // [4, 512]
  const float* vg    = (const float*)d_in[7];  // [4, 1]
  const float* r     = (const float*)d_in[8];  // [1]
  float* out = (float*)d_out;

  float* Kp = (float*)d_ws;                 // [2048, 512] = 4 MB
  float* Qp = Kp + (size_t)2048 * ADIM;     // [256, 512]  = 0.5 MB

  // Key projection: (2048/16 M-tiles) * (512/64 N-groups) = 1024 wave-tiles
  proj_gemm<<<128, 256, 0, stream>>>(key, wk, wkb, Kp, 2048);
  // Query projection: (256/16) * 8 = 128 wave-tiles
  proj_gemm<<<16, 256, 0, stream>>>(query, wq, nullptr, Qp, 256);
  // Fused relu-energy GEMM + mask: one block per (b,q)
  energy_kernel<<<4 * QLEN, 256, 0, stream>>>(Kp, Qp, vv, vg, r, mask, out);
}